// SubmodTriplet_80195629351537
// MI455X (gfx1250) — compile-verified
//
#include <hip/hip_runtime.h>

typedef float v2f __attribute__((ext_vector_type(2)));
typedef float v4f __attribute__((ext_vector_type(4)));
typedef float v8f __attribute__((ext_vector_type(8)));
typedef int   v4i __attribute__((ext_vector_type(4)));

#define BSZ   4096
#define NV    2
#define NTOT  8192      // BSZ * NV
#define DDIM  512
#define NCLS  100
#define INV_T (1.0f / 0.07f)
#define APAD  516       // LDS row stride (floats): 516%64=4 -> conflict-free, 516*4%16=0 -> 16B aligned

#if defined(__AMDGCN__) && __has_builtin(__builtin_amdgcn_global_load_async_to_lds_b128)
#define USE_ASYNC_LDS 1
#else
#define USE_ASYNC_LDS 0
#endif

#if USE_ASYNC_LDS
typedef __attribute__((address_space(1))) v4i GLB_v4i;   // clang prints AS1 as __device__
typedef __attribute__((address_space(3))) v4i LDS_v4i;

__device__ __forceinline__ void async_copy16(const float* g, float* l) {
    __builtin_amdgcn_global_load_async_to_lds_b128(
        (GLB_v4i*)(unsigned long long)(const void*)g,
        (LDS_v4i*)(unsigned int)(unsigned long long)(void*)l,
        0, 0);
}
#endif

// contrast row a -> flat offset into features[bsz][n_views][d]
// contrast[v*BSZ + i] = features[i][v]  =>  off = ((a % BSZ)*NV + a/BSZ) * D
__device__ __forceinline__ int rowOff(int a) {
    return (((a & (BSZ - 1)) << 1) + (a >> 12)) * DDIM;
}

__global__ void hist_zero_kernel(int* __restrict__ hist) {
    int t = threadIdx.x;
    if (t < NCLS) hist[t] = 0;
}

__global__ void hist_kernel(const int* __restrict__ labels, int* __restrict__ hist) {
    int i = blockIdx.x * blockDim.x + threadIdx.x;
    if (i < BSZ) atomicAdd(&hist[labels[i]], 1);
}

// One block = one 32-row anchor strip (A staged in LDS, shared by 4 waves).
// Each wave computes 32x32 output tiles (2x2 register blocking -> 4 independent
// WMMA accumulator chains) over its quarter of the 256 column-tile pairs.
__global__ __launch_bounds__(128)
void supcon_gemm_kernel(const float* __restrict__ feat,
                        const int*   __restrict__ labels,
                        const int*   __restrict__ hist,
                        float*       __restrict__ perAnchor) {
    __shared__ float ldsA[32 * APAD];
    __shared__ float ldsSum[4][32];
    __shared__ float ldsMax[4][32];

    const int rowBase = blockIdx.x << 5;
    const int wave = threadIdx.x >> 5;
    const int lane = threadIdx.x & 31;
    const int m  = lane & 15;   // M (A) / N (B,C) index within a 16-wide tile
    const int kh = lane >> 4;   // half-wave: K pair (A), K parity (B), +8 rows (C)

    // ---- Stage the 32x512 A block into padded LDS (async-to-LDS when available) ----
    #pragma unroll
    for (int i = 0; i < 32; ++i) {
        int c   = i * 128 + threadIdx.x;        // 4096 16-byte chunks
        int row = c >> 7;                       // 128 chunks per 512-float row
        int col = (c & 127) << 2;
        const float* g = feat + rowOff(rowBase + row) + col;
        float*       l = &ldsA[row * APAD + col];
#if USE_ASYNC_LDS
        async_copy16(g, l);
#else
        *(v4f*)l = *(const v4f*)g;
#endif
    }
#if USE_ASYNC_LDS
#if __has_builtin(__builtin_amdgcn_s_wait_asynccnt)
    __builtin_amdgcn_s_wait_asynccnt(0);
#else
    asm volatile("s_wait_asynccnt 0x0" ::: "memory");
#endif
#endif
    __syncthreads();

    // Row labels for the C-rows this lane owns (C layout: vgpr r -> M = r + 8*kh)
    int rowLab[2][8];
    #pragma unroll
    for (int s = 0; s < 2; ++s)
        #pragma unroll
        for (int r = 0; r < 8; ++r)
            rowLab[s][r] = labels[(rowBase + (s << 4) + r + (kh << 3)) & (BSZ - 1)];

    float accSum[2][8], accMax[2][8];
    #pragma unroll
    for (int s = 0; s < 2; ++s)
        #pragma unroll
        for (int r = 0; r < 8; ++r) { accSum[s][r] = 0.0f; accMax[s][r] = -3.4e38f; }

    // A operand per-lane LDS indices (lanes 0-15: M=0..15; vgpr pair = K+2*kh, +1)
    const int aIdx0 = m * APAD + (kh << 1);
    const int aIdx1 = (16 + m) * APAD + (kh << 1);

    const v8f zero8 = {0.f, 0.f, 0.f, 0.f, 0.f, 0.f, 0.f, 0.f};

    for (int jp = wave; jp < NTOT / 32; jp += 4) {
        const int jB0 = jp << 5;
        const int jB1 = jB0 + 16;
        const float* bRow0 = feat + rowOff(jB0 + m);   // 16B-aligned rows
        const float* bRow1 = feat + rowOff(jB1 + m);
        const int colLab0 = labels[(jB0 + m) & (BSZ - 1)];
        const int colLab1 = labels[(jB1 + m) & (BSZ - 1)];

        if (jp + 4 < NTOT / 32) {   // prefetch next tile pair this wave will touch
            __builtin_prefetch(feat + rowOff(jB0 + 128 + m), 0, 1);
            __builtin_prefetch(feat + rowOff(jB1 + 128 + m), 0, 1);
        }

        v8f cc[2][2];
        cc[0][0] = zero8; cc[0][1] = zero8; cc[1][0] = zero8; cc[1][1] = zero8;

        #pragma unroll 4
        for (int k = 0; k < DDIM; k += 4) {
            v2f a0 = *(const v2f*)&ldsA[aIdx0 + k];    // ds_load_b64
            v2f a1 = *(const v2f*)&ldsA[aIdx1 + k];
            v4f x0 = *(const v4f*)(bRow0 + k);         // global_load_b128, 16B aligned
            v4f x1 = *(const v4f*)(bRow1 + k);
            v2f b0, b1;                                // pick K parity for this half-wave
            b0.x = kh ? x0.y : x0.x;  b0.y = kh ? x0.w : x0.z;
            b1.x = kh ? x1.y : x1.x;  b1.y = kh ? x1.w : x1.z;
            // 4 independent accumulator chains keep the matrix pipe busy
            cc[0][0] = __builtin_amdgcn_wmma_f32_16x16x4_f32(false, a0, false, b0, (short)0, cc[0][0], false, false);
            cc[1][0] = __builtin_amdgcn_wmma_f32_16x16x4_f32(false, a1, false, b0, (short)0, cc[1][0], false, false);
            cc[0][1] = __builtin_amdgcn_wmma_f32_16x16x4_f32(false, a0, false, b1, (short)0, cc[0][1], false, false);
            cc[1][1] = __builtin_amdgcn_wmma_f32_16x16x4_f32(false, a1, false, b1, (short)0, cc[1][1], false, false);
        }

        // sim = dot^2 / T ; sign from labels; diagonal excluded from the sum
        #pragma unroll
        for (int s = 0; s < 2; ++s) {
            #pragma unroll
            for (int t = 0; t < 2; ++t) {
                const int jB = t ? jB1 : jB0;
                const int cl = t ? colLab1 : colLab0;
                #pragma unroll
                for (int r = 0; r < 8; ++r) {
                    float d   = cc[s][t][r];
                    float sim = d * d * INV_T;
                    int gi = rowBase + (s << 4) + r + (kh << 3);
                    int gj = jB + m;
                    float sgn = (rowLab[s][r] == cl) ? -1.0f : 1.0f;
                    accSum[s][r] += (gi == gj) ? 0.0f : sgn * sim;
                    accMax[s][r]  = fmaxf(accMax[s][r], sim);  // max includes diagonal
                }
            }
        }
    }

    // Butterfly reduce across the 16 lanes of each half-wave (wave32-safe masks)
    #pragma unroll
    for (int mask = 1; mask <= 8; mask <<= 1) {
        #pragma unroll
        for (int s = 0; s < 2; ++s)
            #pragma unroll
            for (int r = 0; r < 8; ++r) {
                accSum[s][r] += __shfl_xor(accSum[s][r], mask, 32);
                accMax[s][r]  = fmaxf(accMax[s][r], __shfl_xor(accMax[s][r], mask, 32));
            }
    }

    if (m == 0) {   // lane 0 -> rows r+0, lane 16 -> rows r+8 (per strip)
        #pragma unroll
        for (int s = 0; s < 2; ++s)
            #pragma unroll
            for (int r = 0; r < 8; ++r) {
                int idx = (s << 4) + r + (kh << 3);
                ldsSum[wave][idx] = accSum[s][r];
                ldsMax[wave][idx] = accMax[s][r];
            }
    }
    __syncthreads();

    if (threadIdx.x < 32) {
        int t = threadIdx.x;
        float A = ldsSum[0][t] + ldsSum[1][t] + ldsSum[2][t] + ldsSum[3][t];
        float M = fmaxf(fmaxf(ldsMax[0][t], ldsMax[1][t]),
                        fmaxf(ldsMax[2][t], ldsMax[3][t]));
        int gi = rowBase + t;
        int li = labels[gi & (BSZ - 1)];
        // S_i = #neg - #pos = N + 1 - 4*cnt[label_i]   (n_views = 2)
        float S = (float)(NTOT + 1 - 4 * hist[li]);
        perAnchor[gi] = A - M * S;
    }
}

__global__ void reduce_kernel(const float* __restrict__ perAnchor, float* __restrict__ out) {
    __shared__ float s[256];
    float v = 0.0f;
    for (int i = threadIdx.x; i < NTOT; i += 256) v += perAnchor[i];
    s[threadIdx.x] = v;
    __syncthreads();
    for (int off = 128; off > 0; off >>= 1) {
        if (threadIdx.x < off) s[threadIdx.x] += s[threadIdx.x + off];
        __syncthreads();
    }
    if (threadIdx.x == 0) out[0] = -s[0] / (float)NTOT;  // loss = -(T/BT)*mean, T/BT = 1
}

extern "C" void kernel_launch(void* const* d_in, const int* in_sizes, int n_in,
                              void* d_out, int out_size, void* d_ws, size_t ws_size,
                              hipStream_t stream) {
    const float* feat   = (const float*)d_in[0];
    const int*   labels = (const int*)d_in[1];

    float* perAnchor = (float*)d_ws;
    int*   hist      = (int*)((char*)d_ws + NTOT * sizeof(float));

    hist_zero_kernel<<<1, 128, 0, stream>>>(hist);
    hist_kernel<<<(BSZ + 255) / 256, 256, 0, stream>>>(labels, hist);
    supcon_gemm_kernel<<<NTOT / 32, 128, 0, stream>>>(feat, labels, hist, perAnchor);
    reduce_kernel<<<1, 256, 0, stream>>>(perAnchor, (float*)d_out);
}